// GAT_Encoder_82154134438099
// MI455X (gfx1250) — compile-verified
//
#include <hip/hip_runtime.h>
#include <cstdint>

// ---------------- problem constants ----------------
#define NN      100000      // nodes
#define NE      1200000     // edges (before self loops)
#define ET      1300000     // edges + self loops
#define NG      128         // graphs
#define DD      64          // feature dim (in = hid = out)
#define NEG_SLOPE 0.2f

// transposed-W LDS tile: row stride padded 64->68 dwords for conflict-free ds_load_b64
#define WT_STRIDE 68

// CDNA5 async global->LDS staging (ASYNCcnt path). Fallback: plain ds stores.
#ifndef USE_ASYNC_LDS
#define USE_ASYNC_LDS 1
#endif

typedef float    v2f  __attribute__((ext_vector_type(2)));
typedef float    v8f  __attribute__((ext_vector_type(8)));

// monotonic float<->uint encoding so segment-max can use hardware atomicMax(u32)
__device__ __forceinline__ unsigned enc_f(float f) {
  unsigned u = __float_as_uint(f);
  return (u & 0x80000000u) ? ~u : (u | 0x80000000u);
}
__device__ __forceinline__ float dec_f(unsigned u) {
  u = (u & 0x80000000u) ? (u & 0x7fffffffu) : ~u;
  return __uint_as_float(u);
}

// ---------------------------------------------------------------------------
// K1: h = act(in) @ W  (f32 WMMA 16x16x4), fused alpha dot products.
// W is staged once per block into LDS (transposed + padded) via async-to-LDS.
// 1 wave handles 16 rows; 4 waves per block.
// ---------------------------------------------------------------------------
__global__ __launch_bounds__(128)
void gemm_alpha_kernel(const float* __restrict__ in, const float* __restrict__ Wm,
                       const float* __restrict__ avs, const float* __restrict__ avd,
                       float* __restrict__ h, float* __restrict__ asrc,
                       float* __restrict__ adst, int relu_in)
{
  __shared__ float Wt[DD * WT_STRIDE];          // Wt[n*68 + k] = W[k][n]; 17 KB

  // ---- cooperative transposed staging of W (16 KB) ----
  for (int i = threadIdx.x; i < DD * DD; i += 128) {
    const int k = i >> 6;                       // W row  (K index)
    const int n = i & 63;                       // W col  (N index)
#if USE_ASYNC_LDS
    const unsigned dstOff = (unsigned)(uintptr_t)&Wt[n * WT_STRIDE + k];
    const float*   gsrc   = Wm + i;             // coalesced global stream
    asm volatile("global_load_async_to_lds_b32 %0, %1, off"
                 :: "v"(dstOff), "v"(gsrc) : "memory");
#else
    Wt[n * WT_STRIDE + k] = Wm[i];
#endif
  }
#if USE_ASYNC_LDS
  asm volatile("s_wait_asynccnt 0x0" ::: "memory");  // own wave's async loads done
#endif
  __syncthreads();                                   // visible block-wide

  const int lane    = threadIdx.x & 31;
  const int wave    = threadIdx.x >> 5;
  const int rowBase = (blockIdx.x * 4 + wave) * 16;
  if (rowBase >= NN) return;                    // wave-uniform: EXEC all-1s inside
  const int hi = lane >> 4;                     // half-wave select
  const int lo = lane & 15;

  v8f acc[4] = {};                              // 4 N-tiles of 16 cols each
  const float* xrow = in + (size_t)(rowBase + lo) * DD;

  // full-precision f32 WMMA: 16x16x4, K=64 -> 16 k-steps
  #pragma unroll
  for (int kk = 0; kk < 16; ++kk) {
    const int k0 = kk * 4 + hi * 2;             // A layout: VGPR v -> K = k0+v
    float x0 = xrow[k0];
    float x1 = xrow[k0 + 1];
    if (relu_in) { x0 = fmaxf(x0, 0.f); x1 = fmaxf(x1, 0.f); }
    v2f a; a[0] = x0; a[1] = x1;
    #pragma unroll
    for (int t = 0; t < 4; ++t) {
      // B layout: VGPR v -> K = k0+v, N = t*16+lo  ==  Wt[(t*16+lo)*68 + k0 + v]
      const v2f b = *(const v2f*)&Wt[(t * 16 + lo) * WT_STRIDE + k0];  // ds_load_b64, 0-conflict
      acc[t] = __builtin_amdgcn_wmma_f32_16x16x4_f32(
          false, a, false, b, (short)0, acc[t], false, false);
    }
  }

  // store D tiles + fused attention dot products
  const float as0 = avs[lo],      as1 = avs[16 + lo],
              as2 = avs[32 + lo], as3 = avs[48 + lo];
  const float ad0 = avd[lo],      ad1 = avd[16 + lo],
              ad2 = avd[32 + lo], ad3 = avd[48 + lo];
  #pragma unroll
  for (int g = 0; g < 8; ++g) {
    const int r = rowBase + g + hi * 8;         // D layout: VGPR g -> row g+8*hi, col lo
    const float v0 = acc[0][g], v1 = acc[1][g], v2 = acc[2][g], v3 = acc[3][g];
    float* hr = h + (size_t)r * DD;
    hr[lo]      = v0;
    hr[16 + lo] = v1;
    hr[32 + lo] = v2;
    hr[48 + lo] = v3;
    float ps = v0 * as0 + v1 * as1 + v2 * as2 + v3 * as3;
    float pd = v0 * ad0 + v1 * ad1 + v2 * ad2 + v3 * ad3;
    #pragma unroll
    for (int m = 8; m >= 1; m >>= 1) {          // reduce over the 16 lanes of a half
      ps += __shfl_xor(ps, m, 32);
      pd += __shfl_xor(pd, m, 32);
    }
    if (lo == 0) { asrc[r] = ps; adst[r] = pd; }
  }
}

// ---------------------------------------------------------------------------
// init per layer: out = bias (bias folded into accumulator), m_enc=0, denom=0
// ---------------------------------------------------------------------------
__global__ void init_layer_kernel(float* __restrict__ out, const float* __restrict__ bias,
                                  unsigned* __restrict__ menc, float* __restrict__ denom)
{
  const int tid = blockIdx.x * blockDim.x + threadIdx.x;
  if (tid < NN * DD) out[tid] = bias[tid & (DD - 1)];
  if (tid < NN) { menc[tid] = 0u; denom[tid] = 0.f; }
}

// e = leaky_relu(asrc[s] + adst[d]); segment max into m_enc[d]
__global__ void edge_e_kernel(const int* __restrict__ src, const int* __restrict__ dst,
                              const float* __restrict__ asrc, const float* __restrict__ adst,
                              float* __restrict__ ebuf, unsigned* __restrict__ menc)
{
  const int tid = blockIdx.x * blockDim.x + threadIdx.x;
  if (tid >= ET) return;
  int s, d;
  if (tid < NE) { s = src[tid]; d = dst[tid]; } else { s = d = tid - NE; }
  float e = asrc[s] + adst[d];
  e = (e > 0.f) ? e : NEG_SLOPE * e;
  ebuf[tid] = e;
  atomicMax(&menc[d], enc_f(e));
}

// ex = exp(e - m[d]); segment sum into denom[d]
__global__ void edge_exp_kernel(const int* __restrict__ dst, float* __restrict__ ebuf,
                                const unsigned* __restrict__ menc, float* __restrict__ denom)
{
  const int tid = blockIdx.x * blockDim.x + threadIdx.x;
  if (tid >= ET) return;
  const int d = (tid < NE) ? dst[tid] : (tid - NE);
  const float ex = __expf(ebuf[tid] - dec_f(menc[d]));
  ebuf[tid] = ex;
  atomicAdd(&denom[d], ex);
}

// out[d] += (ex/denom[d]) * h[s] ; one wave per edge, 2 cols per lane
__global__ void edge_aggr_kernel(const int* __restrict__ src, const int* __restrict__ dst,
                                 const float* __restrict__ ebuf, const float* __restrict__ denom,
                                 const float* __restrict__ h, float* __restrict__ out)
{
  const long long tid = (long long)blockIdx.x * blockDim.x + threadIdx.x;
  const int edge = (int)(tid >> 5);
  if (edge >= ET) return;
  const int lane = (int)(tid & 31);
  int s, d;
  if (edge < NE) { s = src[edge]; d = dst[edge]; } else { s = d = edge - NE; }
  const float coef = ebuf[edge] / denom[d];
  const float* hr = h + (size_t)s * DD;
  float* orow = out + (size_t)d * DD;
  const int c = lane * 2;
  const float2 hv = *(const float2*)&hr[c];     // global_load_b64
  atomicAdd(&orow[c],     coef * hv.x);
  atomicAdd(&orow[c + 1], coef * hv.y);
}

__global__ void init_pool_kernel(float* __restrict__ pool, float* __restrict__ counts)
{
  const int tid = blockIdx.x * blockDim.x + threadIdx.x;
  if (tid < NG * DD) pool[tid] = 0.f;
  if (tid < NG) counts[tid] = 0.f;
}

// one wave per node, 2 cols per lane
__global__ void pool_kernel(const int* __restrict__ batch, const float* __restrict__ out2,
                            float* __restrict__ pool, float* __restrict__ counts)
{
  const long long tid = (long long)blockIdx.x * blockDim.x + threadIdx.x;
  const int node = (int)(tid >> 5);
  if (node >= NN) return;
  const int lane = (int)(tid & 31);
  const int g = batch[node];
  const int c = lane * 2;
  const float2 hv = *(const float2*)&out2[(size_t)node * DD + c];
  atomicAdd(&pool[(size_t)g * DD + c],     hv.x);
  atomicAdd(&pool[(size_t)g * DD + c + 1], hv.y);
  if (lane == 0) atomicAdd(&counts[g], 1.0f);
}

__global__ void finalize_kernel(const float* __restrict__ pool, const float* __restrict__ counts,
                                float* __restrict__ outp)
{
  const int tid = blockIdx.x * blockDim.x + threadIdx.x;
  if (tid < NG * DD) outp[tid] = pool[tid] / fmaxf(counts[tid >> 6], 1.0f);
}

// ---------------------------------------------------------------------------
extern "C" void kernel_launch(void* const* d_in, const int* in_sizes, int n_in,
                              void* d_out, int out_size, void* d_ws, size_t ws_size,
                              hipStream_t stream)
{
  const float* x    = (const float*)d_in[0];
  const int*   ei   = (const int*)d_in[1];
  const int*   src  = ei;
  const int*   dst  = ei + NE;
  const int*   batch= (const int*)d_in[2];
  const float* W1   = (const float*)d_in[3];
  const float* as1  = (const float*)d_in[4];
  const float* ad1  = (const float*)d_in[5];
  const float* b1   = (const float*)d_in[6];
  const float* W2   = (const float*)d_in[7];
  const float* as2  = (const float*)d_in[8];
  const float* ad2  = (const float*)d_in[9];
  const float* b2   = (const float*)d_in[10];

  // workspace carve (~58 MB)
  float*    A     = (float*)d_ws;            // N x 64  (h of current layer)
  float*    B     = A + (size_t)NN * DD;     // N x 64  (accumulated out of current layer)
  float*    asrc  = B + (size_t)NN * DD;     // N
  float*    adst  = asrc + NN;               // N
  unsigned* menc  = (unsigned*)(adst + NN);  // N
  float*    denom = (float*)(menc + NN);     // N
  float*    ebuf  = denom + NN;              // ET
  float*    pool  = ebuf + ET;               // 128 x 64
  float*    cnts  = pool + NG * DD;          // 128

  const int gemmBlocks = (NN / 16 + 3) / 4;            // 4 waves (64 rows) per block
  const int initBlocks = (NN * DD + 255) / 256;
  const int edgeBlocks = (ET + 255) / 256;
  const int aggrBlocks = (int)(((long long)ET * 32 + 255) / 256);
  const int poolBlocks = (int)(((long long)NN * 32 + 255) / 256);
  const int gBlocks    = (NG * DD + 255) / 256;

  // ---------------- layer 1 ----------------
  gemm_alpha_kernel<<<gemmBlocks, 128, 0, stream>>>(x, W1, as1, ad1, A, asrc, adst, 0);
  init_layer_kernel<<<initBlocks, 256, 0, stream>>>(B, b1, menc, denom);
  edge_e_kernel<<<edgeBlocks, 256, 0, stream>>>(src, dst, asrc, adst, ebuf, menc);
  edge_exp_kernel<<<edgeBlocks, 256, 0, stream>>>(dst, ebuf, menc, denom);
  edge_aggr_kernel<<<aggrBlocks, 256, 0, stream>>>(src, dst, ebuf, denom, A, B);

  // ---------------- layer 2 (ReLU fused into A-load; gemm must read B before re-init)
  gemm_alpha_kernel<<<gemmBlocks, 128, 0, stream>>>(B, W2, as2, ad2, A, asrc, adst, 1);
  init_layer_kernel<<<initBlocks, 256, 0, stream>>>(B, b2, menc, denom);
  edge_e_kernel<<<edgeBlocks, 256, 0, stream>>>(src, dst, asrc, adst, ebuf, menc);
  edge_exp_kernel<<<edgeBlocks, 256, 0, stream>>>(dst, ebuf, menc, denom);
  edge_aggr_kernel<<<aggrBlocks, 256, 0, stream>>>(src, dst, ebuf, denom, A, B);

  // ---------------- global mean pool ----------------
  init_pool_kernel<<<gBlocks, 256, 0, stream>>>(pool, cnts);
  pool_kernel<<<poolBlocks, 256, 0, stream>>>(batch, B, pool, cnts);
  finalize_kernel<<<gBlocks, 256, 0, stream>>>(pool, cnts, (float*)d_out);
}